// GatedLSTM_34557306864148
// MI455X (gfx1250) — compile-verified
//
#include <hip/hip_runtime.h>
#include <hip/hip_bf16.h>
#include <math.h>

// ---------------------------------------------------------------------------
// GatedLSTM for MI455X (gfx1250, wave32).
//   Kernel 1: persistent grid (64 blocks x 256 thr = 512 waves) runs the whole
//             char/word recurrence. Shared input vectors are staged per block
//             into LDS via GLOBAL_LOAD_ASYNC_TO_LDS_B128; char layers are
//             software-pipelined (L1(k) || L0(k+1)) to halve grid barriers.
//   Kernel 2: packs ys into WMMA A-fragment layout (coalesced decoder loads).
//   Kernel 3: decoder GEMM logits = ys @ dec_W^T + b with
//             V_WMMA_F32_16X16X4_F32; 8 M-tiles per wave so dec_W (206 MB)
//             is streamed only 2x.
// ---------------------------------------------------------------------------

#define T_STEPS 256
#define CL      16
#define HW      1024
#define HC      512
#define VW      50257

#define NBLK 64
#define TPB  256
#define NTHR (NBLK * TPB)   // 16384 threads, 512 waves

// workspace layout (floats)
#define YS_OFF    0
#define YS_SZ     (T_STEPS * HW)                 // 262144
#define HCB_OFF   (YS_OFF + YS_SZ)               // [2 parity][2 layer * 512]
#define CC_OFF    (HCB_OFF + 2048)               // [2 layer * 512]
#define HWB_OFF   (CC_OFF + 1024)                // [2 parity][2 layer * 1024]
#define CW_OFF    (HWB_OFF + 4096)               // [2 layer * 1024]
#define XCH_OFF   (CW_OFF + 2048)                // 1024
#define XV_OFF    (XCH_OFF + 1024)               // 1024
#define GV_OFF    (XV_OFF + 1024)                // 1 (+pad)
#define AP_OFF    (GV_OFF + 16)                  // packed A, 262144 floats
#define AP_SZ     (T_STEPS * HW)
#define BAR_BYTE_OFF 2142464                     // 256B aligned, past float area

typedef float v2f __attribute__((ext_vector_type(2)));
typedef float v8f __attribute__((ext_vector_type(8)));
typedef int v4i __attribute__((vector_size(16)));
typedef __attribute__((address_space(1))) v4i gas_v4i;   // global int4
typedef __attribute__((address_space(3))) v4i las_v4i;   // LDS int4

__device__ __forceinline__ float wave_reduce32(float v) {
#pragma unroll
  for (int m = 16; m >= 1; m >>= 1) v += __shfl_xor(v, m, 32);
  return v;
}

__device__ __forceinline__ void gsync(unsigned* bar, unsigned expected) {
  __threadfence();          // release our writes to device scope
  __syncthreads();
  if (threadIdx.x == 0) {
    atomicAdd(bar, 1u);
    while (atomicAdd(bar, 0u) < expected) __builtin_amdgcn_s_sleep(2);
  }
  __syncthreads();
  __threadfence();          // acquire peers' writes
}

// Stage n floats (n % 4 == 0) from global into LDS, block-cooperatively,
// using the CDNA5 async global->LDS path when available.
__device__ __forceinline__ void stage_async(const float* __restrict__ g,
                                            float* __restrict__ s, int n) {
  for (int i = threadIdx.x * 4; i < n; i += TPB * 4) {
#if __has_builtin(__builtin_amdgcn_global_load_async_to_lds_b128)
    __builtin_amdgcn_global_load_async_to_lds_b128(
        (gas_v4i*)(g + i), (las_v4i*)(s + i), 0, 0);
#else
    *(float4*)(s + i) = *(const float4*)(g + i);
#endif
  }
}

__device__ __forceinline__ void stage_wait() {
#if __has_builtin(__builtin_amdgcn_s_wait_asynccnt)
  __builtin_amdgcn_s_wait_asynccnt(0);
#else
  asm volatile("s_wait_asynccnt 0x0" ::: "memory");
#endif
  __syncthreads();
}

__device__ __forceinline__ float sigm(float x) { return 1.f / (1.f + expf(-x)); }
__device__ __forceinline__ float dot4(float4 a, float4 b) {
  return a.x * b.x + a.y * b.y + a.z * b.z + a.w * b.w;
}

// One LSTM cell update for gate-quad j (rows j, H+j, 2H+j, 3H+j). Wave-wide.
// inp/hin point at LDS-staged vectors; weights stream from global (L2-hot).
__device__ __forceinline__ void lstm_j(
    const float* __restrict__ Wih, const float* __restrict__ Whh,
    const float* __restrict__ bih, const float* __restrict__ bhh,
    const float* __restrict__ inp, const float* __restrict__ hin,
    float* __restrict__ hout, float* __restrict__ c,
    float* __restrict__ yout, int H, int j, int lane) {
  const float* wi0 = Wih + (size_t)(0 * H + j) * H;
  const float* wi1 = Wih + (size_t)(1 * H + j) * H;
  const float* wi2 = Wih + (size_t)(2 * H + j) * H;
  const float* wi3 = Wih + (size_t)(3 * H + j) * H;
  const float* wh0 = Whh + (size_t)(0 * H + j) * H;
  const float* wh1 = Whh + (size_t)(1 * H + j) * H;
  const float* wh2 = Whh + (size_t)(2 * H + j) * H;
  const float* wh3 = Whh + (size_t)(3 * H + j) * H;
  float ai = 0.f, af = 0.f, ag = 0.f, ao = 0.f;
  for (int k = lane * 4; k < H; k += 128) {
    float4 x = *(const float4*)(inp + k);
    float4 h = *(const float4*)(hin + k);
    ai += dot4(*(const float4*)(wi0 + k), x) + dot4(*(const float4*)(wh0 + k), h);
    af += dot4(*(const float4*)(wi1 + k), x) + dot4(*(const float4*)(wh1 + k), h);
    ag += dot4(*(const float4*)(wi2 + k), x) + dot4(*(const float4*)(wh2 + k), h);
    ao += dot4(*(const float4*)(wi3 + k), x) + dot4(*(const float4*)(wh3 + k), h);
  }
  ai = wave_reduce32(ai);
  af = wave_reduce32(af);
  ag = wave_reduce32(ag);
  ao = wave_reduce32(ao);
  if (lane == 0) {
    float ig = sigm(ai + bih[j] + bhh[j]);
    float fg = sigm(af + bih[H + j] + bhh[H + j]);
    float gg = tanhf(ag + bih[2 * H + j] + bhh[2 * H + j]);
    float og = sigm(ao + bih[3 * H + j] + bhh[3 * H + j]);
    float cn = fg * c[j] + ig * gg;
    float hn = og * tanhf(cn);
    c[j] = cn;
    hout[j] = hn;
    if (yout) yout[j] = hn;
  }
}

__global__ void __launch_bounds__(TPB, 1) gated_lstm_recurrence(
    const int* __restrict__ xw, const int* __restrict__ xch,
    const float* __restrict__ h_w0, const float* __restrict__ c_w0,
    const float* __restrict__ h_c0, const float* __restrict__ c_c0,
    const float* __restrict__ glove, const float* __restrict__ cemb,
    const float* __restrict__ cWih, const float* __restrict__ cWhh,
    const float* __restrict__ cbih, const float* __restrict__ cbhh,
    const float* __restrict__ wWih, const float* __restrict__ wWhh,
    const float* __restrict__ wbih, const float* __restrict__ wbhh,
    const float* __restrict__ c2e, const float* __restrict__ gw,
    const float* __restrict__ gb, float* __restrict__ ws,
    float* __restrict__ out, unsigned* __restrict__ bar) {
  __shared__ float s_a[HW];   // staged vector A (h0 / xv / cc1)
  __shared__ float s_b[HW];   // staged vector B (h1 / hw)
  __shared__ float s_c[HC];   // staged char-emb row

  const int tid_g = blockIdx.x * TPB + threadIdx.x;
  const int lane = threadIdx.x & 31;
  const int w = (blockIdx.x << 3) + (threadIdx.x >> 5);  // 0..511

  float* ys    = ws + YS_OFF;
  float* hcb   = ws + HCB_OFF;   // [parity][l*512 + j]
  float* ccur  = ws + CC_OFF;    // [l*512 + j]
  float* hwb   = ws + HWB_OFF;   // [parity][l*1024 + j]
  float* cwcur = ws + CW_OFF;    // [l*1024 + j]
  float* xchv  = ws + XCH_OFF;
  float* xv    = ws + XV_OFF;
  float* gval  = ws + GV_OFF;

  const float* cWih1 = cWih + (size_t)4 * HC * HC;
  const float* cWhh1 = cWhh + (size_t)4 * HC * HC;
  const float* wWih1 = wWih + (size_t)4 * HW * HW;
  const float* wWhh1 = wWhh + (size_t)4 * HW * HW;

  // init state from inputs (parity-0 buffers)
  for (int i = tid_g; i < 1024; i += NTHR) { hcb[i] = h_c0[i]; ccur[i] = c_c0[i]; }
  for (int i = tid_g; i < 2048; i += NTHR) { hwb[i] = h_w0[i]; cwcur[i] = c_w0[i]; }

  unsigned ep = 0;
  gsync(bar, (++ep) * NBLK);

  int p0 = 0, p1 = 0, pw = 0;
  for (int t = 0; t < T_STEPS; ++t) {
    // ---- char step 0, layer 0 ----
    {
      const int ci = xch[t * CL];
      stage_async(hcb + p0 * 1024, s_a, HC);
      stage_async(cemb + (size_t)ci * HC, s_c, HC);
      stage_wait();
      lstm_j(cWih, cWhh, cbih, cbhh, s_c, s_a,
             hcb + (1 - p0) * 1024, ccur, nullptr, HC, w, lane);
      p0 ^= 1;
    }
    gsync(bar, (++ep) * NBLK);

    // ---- pipelined: char L1(k) together with char L0(k+1) ----
    for (int kk = 0; kk < CL - 1; ++kk) {
      const int ci = xch[t * CL + kk + 1];
      stage_async(hcb + p0 * 1024, s_a, HC);            // h0 current
      stage_async(hcb + p1 * 1024 + HC, s_b, HC);       // h1 current
      stage_async(cemb + (size_t)ci * HC, s_c, HC);
      stage_wait();
      lstm_j(cWih1, cWhh1, cbih + 4 * HC, cbhh + 4 * HC, s_a, s_b,
             hcb + (1 - p1) * 1024 + HC, ccur + HC, nullptr, HC, w, lane);
      lstm_j(cWih, cWhh, cbih, cbhh, s_c, s_a,
             hcb + (1 - p0) * 1024, ccur, nullptr, HC, w, lane);
      p0 ^= 1;
      p1 ^= 1;
      gsync(bar, (++ep) * NBLK);
    }

    // ---- char L1(last) ----
    stage_async(hcb + p0 * 1024, s_a, HC);
    stage_async(hcb + p1 * 1024 + HC, s_b, HC);
    stage_wait();
    lstm_j(cWih1, cWhh1, cbih + 4 * HC, cbhh + 4 * HC, s_a, s_b,
           hcb + (1 - p1) * 1024 + HC, ccur + HC, nullptr, HC, w, lane);
    p1 ^= 1;
    gsync(bar, (++ep) * NBLK);

    // ---- x_char = c2e @ cc[last] ; g = relu(gw . glove[wi] + gb) ----
    {
      stage_async(ccur + HC, s_a, HC);  // reference uses CELL state of layer 1
      stage_wait();
#pragma unroll
      for (int rep = 0; rep < 2; ++rep) {
        const int j = w + rep * 512;
        const float* row = c2e + (size_t)j * HC;
        float a = 0.f;
        for (int k = lane * 4; k < HC; k += 128)
          a += dot4(*(const float4*)(row + k), *(const float4*)(s_a + k));
        a = wave_reduce32(a);
        if (lane == 0) xchv[j] = a;
      }
      if (w == 0) {
        const int wi = xw[t];
        const float* gl = glove + (size_t)wi * HW;
        float a = 0.f;
        for (int k = lane * 4; k < HW; k += 128)
          a += dot4(*(const float4*)(gw + k), *(const float4*)(gl + k));
        a = wave_reduce32(a);
        if (lane == 0) {
          float g = a + gb[0];
          gval[0] = g > 0.f ? g : 0.f;
        }
      }
    }
    gsync(bar, (++ep) * NBLK);

    // ---- x = (1-g)*x_word + g*x_char ----
    {
      const int wi = xw[t];
      const float g = gval[0];
      const float* gl = glove + (size_t)wi * HW;
      for (int i = tid_g; i < HW; i += NTHR)
        xv[i] = (1.f - g) * gl[i] + g * xchv[i];
    }
    gsync(bar, (++ep) * NBLK);

    // ---- word LSTM layer 0 ----
    stage_async(xv, s_a, HW);
    stage_async(hwb + pw * 2048, s_b, HW);
    stage_wait();
#pragma unroll
    for (int rep = 0; rep < 2; ++rep) {
      const int j = w + rep * 512;
      lstm_j(wWih, wWhh, wbih, wbhh, s_a, s_b,
             hwb + (1 - pw) * 2048, cwcur, nullptr, HW, j, lane);
    }
    gsync(bar, (++ep) * NBLK);

    // ---- word LSTM layer 1 (+ emit ys[t]) ----
    stage_async(hwb + (1 - pw) * 2048, s_a, HW);        // h0 new
    stage_async(hwb + pw * 2048 + HW, s_b, HW);         // h1 current
    stage_wait();
#pragma unroll
    for (int rep = 0; rep < 2; ++rep) {
      const int j = w + rep * 512;
      lstm_j(wWih1, wWhh1, wbih + 4 * HW, wbhh + 4 * HW, s_a, s_b,
             hwb + (1 - pw) * 2048 + HW, cwcur + HW,
             ys + (size_t)t * HW, HW, j, lane);
    }
    gsync(bar, (++ep) * NBLK);
    pw ^= 1;
  }

  // ---- final states -> d_out tail (order: hw, cw, hc, cc) ----
  const size_t OFF = (size_t)T_STEPS * VW;
  for (int i = tid_g; i < 2048; i += NTHR) out[OFF + i] = hwb[pw * 2048 + i];
  for (int i = tid_g; i < 2048; i += NTHR) out[OFF + 2048 + i] = cwcur[i];
  for (int i = tid_g; i < 512; i += NTHR) out[OFF + 4096 + i] = hcb[p0 * 1024 + i];
  for (int i = tid_g; i < 512; i += NTHR)
    out[OFF + 4096 + 512 + i] = hcb[p1 * 1024 + HC + i];
  for (int i = tid_g; i < 1024; i += NTHR) out[OFF + 5120 + i] = ccur[i];
}

// ---------------------------------------------------------------------------
// Pack ys[256,1024] into WMMA A-fragment layout so decoder A loads are one
// coalesced b64 per lane: ap[((m_tile*256 + k/4)*32 + lane)*2 + {0,1}]
//   = ys[m_tile*16 + (lane&15)][k + ((lane>>4)<<1) + {0,1}]
// ---------------------------------------------------------------------------
__global__ void __launch_bounds__(TPB) pack_A(const float* __restrict__ ys,
                                              float* __restrict__ ap) {
  const int idx = blockIdx.x * TPB + threadIdx.x;  // 0 .. 131071
  if (idx >= 16 * 256 * 32) return;
  const int lane = idx & 31;
  const int kq = (idx >> 5) & 255;
  const int mt = idx >> 13;
  const int M = mt * 16 + (lane & 15);
  const int k = kq * 4 + ((lane >> 4) << 1);
  float2 val;
  val.x = ys[(size_t)M * HW + k];
  val.y = ys[(size_t)M * HW + k + 1];
  *(float2*)(ap + (size_t)idx * 2) = val;
}

// ---------------------------------------------------------------------------
// Decoder: logits[256, 50257] = ys @ dec_W^T + b.
// One wave owns a 128(M) x 16(N) strip (8 WMMA tiles) so dec_W is streamed 2x.
// B fragment: lane = {W[v][k+kh], W[v][k+kh+1]}, v = n_tile*16 + (lane&15),
// kh = (lane>>4)*2 -- mirrors the documented f32 A layout with N<->M roles.
// ---------------------------------------------------------------------------
#define N_TILES ((VW + 15) / 16)            // 3142
#define DEC_MG 2                            // two 128-row M groups
#define DEC_TASKS (N_TILES * DEC_MG)        // 6284 wave tasks
#define DEC_BLOCKS ((DEC_TASKS + 7) / 8)    // 786

__global__ void __launch_bounds__(TPB) decoder_gemm_wmma(
    const float* __restrict__ ap, const float* __restrict__ W,
    const float* __restrict__ b, float* __restrict__ out) {
  const int lane = threadIdx.x & 31;
  const int task = blockIdx.x * 8 + (threadIdx.x >> 5);
  if (task >= DEC_TASKS) return;
  const int n_tile = task % N_TILES;
  const int m_grp = task / N_TILES;          // 0..1, m_tiles 8g..8g+7
  const int kh = (lane >> 4) << 1;
  int v = n_tile * 16 + (lane & 15);
  const int vc = v < VW ? v : (VW - 1);
  const float* wrow = W + (size_t)vc * HW;

  v8f z;
#pragma unroll
  for (int i = 0; i < 8; ++i) z[i] = 0.f;
  v8f acc[8] = {z, z, z, z, z, z, z, z};

  for (int kq = 0; kq < 256; ++kq) {
    const int k = kq * 4;
    if ((kq & 7) == 0) __builtin_prefetch(wrow + k + 128, 0, 1);
    v2f bb;
    bb.x = wrow[k + kh];
    bb.y = wrow[k + kh + 1];
#pragma unroll
    for (int mi = 0; mi < 8; ++mi) {
      const int mt = m_grp * 8 + mi;
      v2f aa = *(const v2f*)(ap + ((size_t)(mt * 256 + kq) * 32 + lane) * 2);
      acc[mi] = __builtin_amdgcn_wmma_f32_16x16x4_f32(
          false, aa, false, bb, (short)0, acc[mi], false, false);
    }
  }

  if (v < VW) {
    const float bias = b[v];
    const int rbase = (lane >> 4) << 3;      // lanes 16-31 hold M=8..15
#pragma unroll
    for (int mi = 0; mi < 8; ++mi) {
#pragma unroll
      for (int r = 0; r < 8; ++r) {
        const int row = (m_grp * 8 + mi) * 16 + rbase + r;
        out[(size_t)row * VW + v] = acc[mi][r] + bias;
      }
    }
  }
}

extern "C" void kernel_launch(void* const* d_in, const int* in_sizes, int n_in,
                              void* d_out, int out_size, void* d_ws, size_t ws_size,
                              hipStream_t stream) {
  (void)in_sizes; (void)n_in; (void)out_size; (void)ws_size;
  const int*   xw    = (const int*)d_in[0];
  const int*   xch   = (const int*)d_in[1];
  const float* h_w0  = (const float*)d_in[2];
  const float* c_w0  = (const float*)d_in[3];
  const float* h_c0  = (const float*)d_in[4];
  const float* c_c0  = (const float*)d_in[5];
  const float* glove = (const float*)d_in[6];
  const float* cemb  = (const float*)d_in[7];
  const float* cWih  = (const float*)d_in[8];
  const float* cWhh  = (const float*)d_in[9];
  const float* cbih  = (const float*)d_in[10];
  const float* cbhh  = (const float*)d_in[11];
  const float* wWih  = (const float*)d_in[12];
  const float* wWhh  = (const float*)d_in[13];
  const float* wbih  = (const float*)d_in[14];
  const float* wbhh  = (const float*)d_in[15];
  const float* c2e   = (const float*)d_in[16];
  const float* gw    = (const float*)d_in[17];
  const float* gb    = (const float*)d_in[18];
  const float* decW  = (const float*)d_in[19];
  const float* decb  = (const float*)d_in[20];

  float*    ws  = (float*)d_ws;
  unsigned* bar = (unsigned*)((char*)d_ws + BAR_BYTE_OFF);
  float*    out = (float*)d_out;

  // reset grid barrier (graph-capturable memset node)
  (void)hipMemsetAsync((char*)d_ws + BAR_BYTE_OFF, 0, 256, stream);

  gated_lstm_recurrence<<<NBLK, TPB, 0, stream>>>(
      xw, xch, h_w0, c_w0, h_c0, c_c0, glove, cemb,
      cWih, cWhh, cbih, cbhh, wWih, wWhh, wbih, wbhh,
      c2e, gw, gb, ws, out, bar);

  pack_A<<<512, TPB, 0, stream>>>(ws + YS_OFF, ws + AP_OFF);

  decoder_gemm_wmma<<<DEC_BLOCKS, TPB, 0, stream>>>(ws + AP_OFF, decW, decb, out);
}